// OrthogonalLoss_44272522887721
// MI455X (gfx1250) — compile-verified
//
#include <hip/hip_runtime.h>
#include <hip/hip_bf16.h>
#include <math.h>

typedef __attribute__((ext_vector_type(2))) float v2f;
typedef __attribute__((ext_vector_type(8))) float v8f;

#define COS_EPS 1e-8f

// ---------------------------------------------------------------------------
// Kernel 1: zero the 3 label counters in workspace (harness poisons d_ws).
// ---------------------------------------------------------------------------
__global__ void ol_zero_counts(unsigned int* __restrict__ cnt) {
    if (threadIdx.x < 4) cnt[threadIdx.x] = 0u;
}

// ---------------------------------------------------------------------------
// Kernel 2: histogram of labels (values 0..2). LDS-local counts, then 3
// global integer atomics per block (deterministic).
// ---------------------------------------------------------------------------
__global__ void ol_label_hist(const int* __restrict__ labels, int n,
                              unsigned int* __restrict__ cnt) {
    __shared__ unsigned int lc[3];
    if (threadIdx.x < 3) lc[threadIdx.x] = 0u;
    __syncthreads();

    int idx    = blockIdx.x * blockDim.x + threadIdx.x;
    int stride = gridDim.x * blockDim.x;
    for (int i = idx; i < n; i += stride) {
        int v = labels[i];
        if (v >= 0 && v < 3) atomicAdd(&lc[v], 1u);
    }
    __syncthreads();
    if (threadIdx.x < 3) atomicAdd(&cnt[threadIdx.x], lc[threadIdx.x]);
}

// ---------------------------------------------------------------------------
// Kernel 3 (single wave of 32): Gram matrix of the six vectors
//   rows 0..2 : tensor_1[0..2]   rows 3..5 : tensor_2[0..2]
// via V_WMMA_F32_16X16X4_F32 accumulating over K in chunks of 4.
// A-tile layout (32-bit A 16x4): m = lane&15, k = 2*(lane>>4) + vgpr.
// B-tile 4x16 is lane-symmetric (n = lane&15), so the same register pair
// serves as both A and B -> D accumulates G[m][n] = dot(vec_m, vec_n).
//
// Rows 6..15 are fed from tensor_1 row 0 (any finite data works): a Gram
// entry G[m][n] depends only on rows m,n, and we never read m,n > 5. This
// keeps control flow fully uniform -> no EXEC save/restore in the loop.
//
// C/D layout: lanes 0..15 hold N=lane, VGPR v holds M=v (for M<8).
// ---------------------------------------------------------------------------
__global__ void ol_gram_wmma(const float* __restrict__ t1,
                             const float* __restrict__ t2,
                             const unsigned int* __restrict__ cnt,
                             int d, int n_rows,
                             float* __restrict__ out) {
    const int lane  = threadIdx.x;        // 0..31, one wave
    const int m     = lane & 15;          // matrix row this lane feeds
    const int khalf = (lane >> 4) << 1;   // 0 (lanes 0-15) or 2 (lanes 16-31)

    // Always-valid per-lane row pointer (rows >=6 are don't-care data).
    const float* rowp;
    if (m < 3)      rowp = t1 + (size_t)m * d;
    else if (m < 6) rowp = t2 + (size_t)(m - 3) * d;
    else            rowp = t1;            // row 0: harmless filler
    rowp += khalf;

    // Two independent accumulators to break the WMMA C->C dependency chain.
    v8f acc0 = {0.f, 0.f, 0.f, 0.f, 0.f, 0.f, 0.f, 0.f};
    v8f acc1 = {0.f, 0.f, 0.f, 0.f, 0.f, 0.f, 0.f, 0.f};

    for (int base = 0; base < d; base += 8) {
        v2f ab0 = *(const v2f*)(rowp + base);       // K-chunk c   (8B aligned)
        v2f ab1 = *(const v2f*)(rowp + base + 4);   // K-chunk c+1
        acc0 = __builtin_amdgcn_wmma_f32_16x16x4_f32(
            false, ab0, false, ab0, (short)0, acc0, false, false);
        acc1 = __builtin_amdgcn_wmma_f32_16x16x4_f32(
            false, ab1, false, ab1, (short)0, acc1, false, false);
    }

#pragma unroll
    for (int v = 0; v < 8; ++v) acc0[v] += acc1[v];

    // Spill G rows 0..7 (we need rows 0..5) to LDS for lane 0.
    __shared__ float G[8][16];
    if (lane < 16) {
#pragma unroll
        for (int v = 0; v < 8; ++v) G[v][lane] = acc0[v];
    }
    __syncthreads();

    if (lane == 0) {
        float total = 0.f;
#pragma unroll
        for (int r = 0; r < 3; ++r) {
            float dotv = G[r][3 + r];               // dot(t1[r], t2[r])
            float na   = sqrtf(G[r][r]);            // ||t1[r]||
            float nb   = sqrtf(G[3 + r][3 + r]);    // ||t2[r]||
            float cosv = dotv / fmaxf(na * nb, COS_EPS);
            float c    = (float)cnt[r];
            // labels in {0,2} -> (1 - cos); label == 1 -> |cos|
            total += (r == 1) ? c * fabsf(cosv) : c * (1.f - cosv);
        }
        out[0] = total / (float)n_rows;
    }
}

// ---------------------------------------------------------------------------
// Launcher
// ---------------------------------------------------------------------------
extern "C" void kernel_launch(void* const* d_in, const int* in_sizes, int n_in,
                              void* d_out, int out_size, void* d_ws, size_t ws_size,
                              hipStream_t stream) {
    const float* t1     = (const float*)d_in[0];
    const float* t2     = (const float*)d_in[1];
    const int*   labels = (const int*)d_in[2];   // harness delivers integer inputs as int32
    float*       out    = (float*)d_out;

    const int n_labels = in_sizes[2];                 // 65536
    const int d        = in_sizes[0] / n_labels;      // 512 (N == B here)
    const int n_rows   = n_labels;                    // divisor = tensor_1.shape[0]

    unsigned int* cnt = (unsigned int*)d_ws;

    ol_zero_counts<<<1, 32, 0, stream>>>(cnt);

    int threads = 256;
    int blocks  = (n_labels + threads - 1) / threads;
    if (blocks > 512) blocks = 512;
    ol_label_hist<<<blocks, threads, 0, stream>>>(labels, n_labels, cnt);

    ol_gram_wmma<<<1, 32, 0, stream>>>(t1, t2, cnt, d, n_rows, out);
}